// GATEncoder_84894323572904
// MI455X (gfx1250) — compile-verified
//
#include <hip/hip_runtime.h>
#include <cstdint>
#include <cstddef>

#define NN   50000
#define EE   800000
#define INC  128
#define EDC  16
#define HH   4
#define HIDC 64
#define OUTC 128
#define GG   64

typedef float v2f __attribute__((ext_vector_type(2)));
typedef float v8f __attribute__((ext_vector_type(8)));

// ---------------- utility ----------------

__global__ void fillf(float* __restrict__ p, float v, long long n) {
    long long i = (long long)blockIdx.x * blockDim.x + threadIdx.x;
    long long stride = (long long)gridDim.x * blockDim.x;
    for (; i < n; i += stride) p[i] = v;
}

__device__ __forceinline__ void atomicMaxF(float* addr, float v) {
    // sign-aware ordered-int trick; works for all sign combinations.
    if (v >= 0.0f) atomicMax((int*)addr, __float_as_int(v));
    else           atomicMin((unsigned int*)addr, __float_as_uint(v));
}

// ---------------- precompute ----------------

// partial sums of edge_attr columns (for self-loop mean fill)
__global__ void mean_ea_k(const float* __restrict__ ea, float* __restrict__ msum) {
    float acc[EDC];
#pragma unroll
    for (int d = 0; d < EDC; d++) acc[d] = 0.0f;
    long long i = (long long)blockIdx.x * blockDim.x + threadIdx.x;
    long long stride = (long long)gridDim.x * blockDim.x;
    for (long long e = i; e < EE; e += stride) {
        const float* p = ea + e * EDC;
#pragma unroll
        for (int d = 0; d < EDC; d++) acc[d] += p[d];
    }
#pragma unroll
    for (int d = 0; d < EDC; d++) atomicAdd(&msum[d], acc[d]);
}

// We_fused[d][hd] = sum_cc We[d, hd*c+cc] * ae[hd,cc]; ael[hd] = mean_ea . We_fused
__global__ void wef_k(const float* __restrict__ We, const float* __restrict__ ae,
                      const float* __restrict__ msum, float* __restrict__ Wf,
                      float* __restrict__ ael, int c, int HK) {
    __shared__ float sWf[EDC * HH];
    int tid = threadIdx.x;              // 0..63
    int d = tid & 15, hd = tid >> 4;
    const float* wp = We + (size_t)d * HK + hd * c;
    const float* ap = ae + hd * c;
    float s = 0.0f;
    for (int cc = 0; cc < c; cc++) s += wp[cc] * ap[cc];
    Wf[d * HH + hd] = s;
    sWf[d * HH + hd] = s;
    __syncthreads();
    if (tid < HH) {
        const float invE = 1.0f / (float)EE;
        float t = 0.0f;
        for (int dd = 0; dd < EDC; dd++) t += (msum[dd] * invE) * sWf[dd * HH + tid];
        ael[tid] = t;
    }
}

// a_e[e][hd] = ea[e,:] @ We_fused[:,hd]
__global__ __launch_bounds__(256) void a_e_k(const float* __restrict__ ea,
                                             const float* __restrict__ Wf,
                                             float* __restrict__ a_e) {
    __shared__ float sWf[EDC * HH];
    if (threadIdx.x < EDC * HH) sWf[threadIdx.x] = Wf[threadIdx.x];
    __syncthreads();
    long long e = (long long)blockIdx.x * blockDim.x + threadIdx.x;
    if (e >= EE) return;
    const float* p = ea + e * EDC;
    float acc[HH] = {0.f, 0.f, 0.f, 0.f};
#pragma unroll
    for (int d = 0; d < EDC; d++) {
        float v = p[d];
#pragma unroll
        for (int hd = 0; hd < HH; hd++) acc[hd] += v * sWf[d * HH + hd];
    }
#pragma unroll
    for (int hd = 0; hd < HH; hd++) a_e[e * HH + hd] = acc[hd];
}

// ---------------- dense GEMM via fp32 WMMA ----------------
// C[NN x KO] = X[NN x K] @ W[K x KO]; one wave -> 16 rows x 64 cols of C.
__global__ __launch_bounds__(128) void gemm_wmma(
    const float* __restrict__ X, const float* __restrict__ W,
    float* __restrict__ Hout, int rowTiles, int K, int KO) {
    int lane = threadIdx.x;                       // 0..31
    int rowTile = blockIdx.x * 4 + threadIdx.y;   // wave-uniform
    if (rowTile >= rowTiles) return;
    int colBase = blockIdx.y * 64;
    int r = lane & 15, half = lane >> 4;

    v8f acc0 = {}, acc1 = {}, acc2 = {}, acc3 = {};
    const float* xrow = X + (size_t)(rowTile * 16 + r) * K;
    const float* wc = W + colBase + r;

    for (int k = 0; k < K; k += 4) {
        int ka = k + 2 * half;
        // A 16x4 fp32: vgpr0 = K {0|2}, vgpr1 = K {1|3} for lane halves
        v2f a = { xrow[ka], xrow[ka + 1] };
        const float* w0 = wc + (size_t)ka * KO;
        const float* w1 = wc + (size_t)(ka + 1) * KO;
        v2f b0 = { w0[0],  w1[0]  };
        v2f b1 = { w0[16], w1[16] };
        v2f b2 = { w0[32], w1[32] };
        v2f b3 = { w0[48], w1[48] };
        acc0 = __builtin_amdgcn_wmma_f32_16x16x4_f32(false, a, false, b0, (short)0, acc0, false, false);
        acc1 = __builtin_amdgcn_wmma_f32_16x16x4_f32(false, a, false, b1, (short)0, acc1, false, false);
        acc2 = __builtin_amdgcn_wmma_f32_16x16x4_f32(false, a, false, b2, (short)0, acc2, false, false);
        acc3 = __builtin_amdgcn_wmma_f32_16x16x4_f32(false, a, false, b3, (short)0, acc3, false, false);
    }
    // D layout: vgpr i -> row m = i + 8*half, col n = r
#pragma unroll
    for (int i = 0; i < 8; i++) {
        size_t row = (size_t)(rowTile * 16 + i + 8 * half);
        float* o = Hout + row * KO + colBase + r;
        o[0]  = acc0[i];
        o[16] = acc1[i];
        o[32] = acc2[i];
        o[48] = acc3[i];
    }
}

// ---------------- attention logits per node ----------------
__global__ void as_ad_k(const float* __restrict__ h, const float* __restrict__ asrc,
                        const float* __restrict__ adst, float* __restrict__ a_s,
                        float* __restrict__ a_d, int c, int HK) {
    int idx = blockIdx.x * blockDim.x + threadIdx.x;
    if (idx >= NN * HH) return;
    int n = idx >> 2, hd = idx & 3;
    const float* hp = h + (size_t)n * HK + hd * c;
    const float* ap = asrc + hd * c;
    const float* dp = adst + hd * c;
    float s = 0.0f, t = 0.0f;
    for (int cc = 0; cc < c; cc++) { float v = hp[cc]; s += v * ap[cc]; t += v * dp[cc]; }
    a_s[idx] = s;
    a_d[idx] = t;
}

// ---------------- edge pass 1: segment max ----------------
__global__ void edge_amax(const int* __restrict__ src, const int* __restrict__ dst,
                          const float* __restrict__ a_s, const float* __restrict__ a_d,
                          const float* __restrict__ a_e, const float* __restrict__ ael,
                          float* __restrict__ amax) {
    long long idx = (long long)blockIdx.x * blockDim.x + threadIdx.x;
    if (idx >= (long long)(EE + NN) * HH) return;
    long long e = idx >> 2;
    int hd = (int)(idx & 3);
    int s, d; float ae;
    if (e < EE) { s = src[e]; d = dst[e]; ae = a_e[idx]; }
    else        { s = d = (int)(e - EE); ae = ael[hd]; }
    float v = a_s[s * HH + hd] + a_d[d * HH + hd] + ae;
    v = v > 0.0f ? v : 0.2f * v;   // leaky relu
    atomicMaxF(&amax[d * HH + hd], v);
}

// ---------------- edge pass 2: exp + denom + weighted scatter ----------------
// one wave per edge; lanes stripe the HK feature columns.
__global__ __launch_bounds__(256) void edge_acc(
    const int* __restrict__ src, const int* __restrict__ dst,
    const float* __restrict__ a_s, const float* __restrict__ a_d,
    const float* __restrict__ a_e, const float* __restrict__ ael,
    const float* __restrict__ amax, float* __restrict__ denom,
    const float* __restrict__ h, float* __restrict__ U,
    int HK, int cshift) {
    int lane = threadIdx.x;
    long long e = (long long)blockIdx.x * 8 + threadIdx.y;
    if (e >= (long long)(EE + NN)) return;
    int s, d; float aev[HH];
    if (e < EE) {
        s = src[e]; d = dst[e];
#pragma unroll
        for (int hd = 0; hd < HH; hd++) aev[hd] = a_e[e * HH + hd];
    } else {
        s = d = (int)(e - EE);
#pragma unroll
        for (int hd = 0; hd < HH; hd++) aev[hd] = ael[hd];
    }
    float ex[HH];
#pragma unroll
    for (int hd = 0; hd < HH; hd++) {
        float v = a_s[s * HH + hd] + a_d[d * HH + hd] + aev[hd];
        v = v > 0.0f ? v : 0.2f * v;
        ex[hd] = expf(v - amax[d * HH + hd]);
    }
    if (lane == 0) {
#pragma unroll
        for (int hd = 0; hd < HH; hd++) atomicAdd(&denom[d * HH + hd], ex[hd]);
    }
    const float* hs = h + (size_t)s * HK;
    float* ud = U + (size_t)d * HK;
    for (int j = lane; j < HK; j += 32)
        atomicAdd(&ud[j], hs[j] * ex[j >> cshift]);
}

// ---------------- finalize ----------------
__global__ void fin_concat(float* __restrict__ U, const float* __restrict__ denom,
                           const float* __restrict__ b, int HK, int cshift) {
    long long idx = (long long)blockIdx.x * blockDim.x + threadIdx.x;
    if (idx >= (long long)NN * HK) return;
    int n = (int)(idx / HK);
    int j = (int)(idx % HK);
    int hd = j >> cshift;
    float v = U[idx] / (denom[n * HH + hd] + 1e-16f) + b[j];
    v = v > 0.0f ? v : (expf(v) - 1.0f);   // ELU
    U[idx] = v;
}

__global__ void fin_mean(const float* __restrict__ U, const float* __restrict__ denom,
                         const float* __restrict__ b, float* __restrict__ out) {
    int idx = blockIdx.x * blockDim.x + threadIdx.x;
    if (idx >= NN * OUTC) return;
    int n = idx / OUTC, cc = idx % OUTC;
    float s = 0.0f;
#pragma unroll
    for (int hd = 0; hd < HH; hd++)
        s += U[(size_t)n * (HH * OUTC) + hd * OUTC + cc] / (denom[n * HH + hd] + 1e-16f);
    out[idx] = s * 0.25f + b[cc];
}

// ---------------- global mean pool ----------------
__global__ void pool_k(const float* __restrict__ hf, const int* __restrict__ batch,
                       float* __restrict__ gout) {
    int idx = blockIdx.x * blockDim.x + threadIdx.x;
    if (idx >= NN * OUTC) return;
    int n = idx / OUTC, cc = idx % OUTC;
    atomicAdd(&gout[batch[n] * OUTC + cc], hf[idx]);
}

__global__ void cnt_k(const int* __restrict__ batch, float* __restrict__ cnt) {
    int n = blockIdx.x * blockDim.x + threadIdx.x;
    if (n >= NN) return;
    atomicAdd(&cnt[batch[n]], 1.0f);
}

__global__ void div_k(float* __restrict__ gout, const float* __restrict__ cnt) {
    int idx = blockIdx.x * blockDim.x + threadIdx.x;
    if (idx >= GG * OUTC) return;
    gout[idx] /= fmaxf(cnt[idx / OUTC], 1.0f);
}

// ---------------- orchestration ----------------
extern "C" void kernel_launch(void* const* d_in, const int* in_sizes, int n_in,
                              void* d_out, int out_size, void* d_ws, size_t ws_size,
                              hipStream_t stream) {
    (void)in_sizes; (void)n_in; (void)out_size; (void)ws_size;
    const float* x    = (const float*)d_in[0];
    const int*   ei   = (const int*)d_in[1];
    const int*   srcI = ei;
    const int*   dstI = ei + EE;
    const float* ea   = (const float*)d_in[2];
    const int*   batch = (const int*)d_in[3];

    struct Lp { const float *W, *asrc, *adst, *We, *ae, *b; int cin, c; };
    Lp L[3];
    for (int i = 0; i < 3; i++) {
        L[i].W    = (const float*)d_in[4 + 6 * i];
        L[i].asrc = (const float*)d_in[5 + 6 * i];
        L[i].adst = (const float*)d_in[6 + 6 * i];
        L[i].We   = (const float*)d_in[7 + 6 * i];
        L[i].ae   = (const float*)d_in[8 + 6 * i];
        L[i].b    = (const float*)d_in[9 + 6 * i];
    }
    L[0].cin = INC;       L[0].c = HIDC;
    L[1].cin = HH * HIDC; L[1].c = HIDC;
    L[2].cin = HH * HIDC; L[2].c = OUTC;

    float* ws    = (float*)d_ws;
    float* Buf1  = ws;                               // N x 512 (h)
    float* Buf2  = Buf1 + (size_t)NN * 512;          // N x 512 (U / next act)
    float* a_s   = Buf2 + (size_t)NN * 512;
    float* a_d   = a_s + (size_t)NN * HH;
    float* amax  = a_d + (size_t)NN * HH;
    float* denom = amax + (size_t)NN * HH;
    float* a_e   = denom + (size_t)NN * HH;          // E x H
    float* ael   = a_e + (size_t)EE * HH;            // H
    float* Wf    = ael + HH;                         // ED x H
    float* msum  = Wf + EDC * HH;                    // ED
    float* cnt   = msum + EDC;                       // G

    fillf<<<1, 256, 0, stream>>>(msum, 0.0f, EDC);
    mean_ea_k<<<256, 256, 0, stream>>>(ea, msum);

    const float* actIn = x;
    const int rowTiles = (NN + 15) / 16;   // 3125 (exact)
    for (int l = 0; l < 3; l++) {
        int c = L[l].c, HK = HH * c, K = L[l].cin;
        int cshift = (c == 64) ? 6 : 7;

        wef_k<<<1, 64, 0, stream>>>(L[l].We, L[l].ae, msum, Wf, ael, c, HK);
        a_e_k<<<(EE + 255) / 256, 256, 0, stream>>>(ea, Wf, a_e);

        dim3 g((rowTiles + 3) / 4, HK / 64), blk(32, 4);
        gemm_wmma<<<g, blk, 0, stream>>>(actIn, L[l].W, Buf1, rowTiles, K, HK);

        as_ad_k<<<(NN * HH + 255) / 256, 256, 0, stream>>>(Buf1, L[l].asrc, L[l].adst, a_s, a_d, c, HK);

        fillf<<<256, 256, 0, stream>>>(amax, -3.0e38f, (long long)NN * HH);
        fillf<<<256, 256, 0, stream>>>(denom, 0.0f, (long long)NN * HH);
        fillf<<<1024, 256, 0, stream>>>(Buf2, 0.0f, (long long)NN * HK);

        long long tot = (long long)(EE + NN) * HH;
        edge_amax<<<(unsigned)((tot + 255) / 256), 256, 0, stream>>>(srcI, dstI, a_s, a_d, a_e, ael, amax);
        edge_acc<<<(EE + NN + 7) / 8, dim3(32, 8), 0, stream>>>(srcI, dstI, a_s, a_d, a_e, ael,
                                                                amax, denom, Buf1, Buf2, HK, cshift);
        if (l < 2) {
            fin_concat<<<(unsigned)(((long long)NN * HK + 255) / 256), 256, 0, stream>>>(Buf2, denom, L[l].b, HK, cshift);
            actIn = Buf2;
        } else {
            fin_mean<<<(NN * OUTC + 255) / 256, 256, 0, stream>>>(Buf2, denom, L[l].b, Buf1);
        }
    }

    float* gout = (float*)d_out;
    fillf<<<(GG * OUTC + 255) / 256, 256, 0, stream>>>(gout, 0.0f, GG * OUTC);
    fillf<<<1, 256, 0, stream>>>(cnt, 0.0f, GG);
    pool_k<<<(NN * OUTC + 255) / 256, 256, 0, stream>>>(Buf1, batch, gout);
    cnt_k<<<(NN + 255) / 256, 256, 0, stream>>>(batch, cnt);
    div_k<<<(GG * OUTC + 255) / 256, 256, 0, stream>>>(gout, cnt);
}